// RNNAttentionLanguageModel_22454089024206
// MI455X (gfx1250) — compile-verified
//
#include <hip/hip_runtime.h>
#include <hip/hip_bf16.h>
#include <math.h>

typedef __attribute__((ext_vector_type(2))) float v2f;
typedef __attribute__((ext_vector_type(8))) float v8f;

#define VOCAB 32000
#define EMB   256
#define HID   256
#define BB    2
#define TT    512
#define MROWS (BB * TT)   // 1024
#define G4    (4 * HID)   // 1024

// ---------------------------------------------------------------------------
// Embedding gather: xe[m, e] = emb[x[m], e]
// ---------------------------------------------------------------------------
__global__ void k_embed(const int* __restrict__ x, const float* __restrict__ emb,
                        float* __restrict__ xe) {
  int m = blockIdx.x;          // 0..1023
  int e = threadIdx.x;         // 0..255
  int tok = x[m];
  xe[(size_t)m * EMB + e] = emb[(size_t)tok * EMB + e];
}

// ---------------------------------------------------------------------------
// Generic fp32 WMMA GEMM:  C(M,N) = A(M,K) @ W(N,K)^T + bias1[n] (+ bias2[n])
// Register-blocked: each wave computes a 4(M) x 2(N) block of 16x16 tiles
// (8 v8f accumulators). Per K-step: 6 b64 loads feed 8 v_wmma_f32_16x16x4_f32.
// A layout (32-bit 16x4): lanes 0-15 rows M, VGPR0/1 = K0/K1; lanes 16-31 K2/K3.
// B (4x16 = W^T tile) mirrored: lane n holds column, K = 2*half + vgpr.
// ---------------------------------------------------------------------------
#define MBLK 4
#define NBLK 2

__global__ void k_gemm_f32_wmma(const float* __restrict__ A, const float* __restrict__ W,
                                const float* __restrict__ bias1, const float* __restrict__ bias2,
                                float* __restrict__ C, int M, int N, int K) {
  int wave = (int)((blockIdx.x * blockDim.x + threadIdx.x) >> 5);
  int lane = threadIdx.x & 31;
  int nst   = N / (16 * NBLK);
  int tiles = (M / (16 * MBLK)) * nst;
  if (wave >= tiles) return;               // uniform per wave
  int tm = wave / nst, tn = wave % nst;
  int m0 = tm * (16 * MBLK), n0 = tn * (16 * NBLK);
  int half = lane >> 4, l16 = lane & 15;

  const float* ap = A + (size_t)(m0 + l16) * K + 2 * half;
  const float* wp = W + (size_t)(n0 + l16) * K + 2 * half;

  v8f acc[MBLK][NBLK] = {};
  for (int k0 = 0; k0 < K; k0 += 4) {
    v2f a[MBLK], b[NBLK];
#pragma unroll
    for (int i = 0; i < MBLK; ++i)
      a[i] = *(const v2f*)(ap + (size_t)(16 * i) * K + k0);
#pragma unroll
    for (int j = 0; j < NBLK; ++j)
      b[j] = *(const v2f*)(wp + (size_t)(16 * j) * K + k0);
    __builtin_prefetch(wp + k0 + 128, 0, 1);   // global_prefetch_b8, stream W
#pragma unroll
    for (int i = 0; i < MBLK; ++i)
#pragma unroll
      for (int j = 0; j < NBLK; ++j)
        acc[i][j] = __builtin_amdgcn_wmma_f32_16x16x4_f32(
            /*neg_a=*/false, a[i], /*neg_b=*/false, b[j],
            /*c_mod=*/(short)0, acc[i][j], /*reuse_a=*/false, /*reuse_b=*/false);
  }

  float bv[NBLK];
#pragma unroll
  for (int j = 0; j < NBLK; ++j) {
    float v = 0.0f;
    if (bias1) v += bias1[n0 + 16 * j + l16];
    if (bias2) v += bias2[n0 + 16 * j + l16];
    bv[j] = v;
  }

  // C/D layout: VGPR p -> row M = p (lanes 0-15) or p+8 (lanes 16-31); col = l16
#pragma unroll
  for (int i = 0; i < MBLK; ++i)
#pragma unroll
    for (int j = 0; j < NBLK; ++j)
#pragma unroll
      for (int p = 0; p < 8; ++p) {
        int row = m0 + 16 * i + p + 8 * half;
        C[(size_t)row * N + n0 + 16 * j + l16] = acc[i][j][p] + bv[j];
      }
}

// ---------------------------------------------------------------------------
// LSTM: 8 workgroups, each keeps a 128KB K-major slice of W_hh in LDS
// (4 gates x 32 h-indices x 256 K x fp32). Hidden state exchanged through L2
// (outputs rows) with an agent-scope atomic arrive counter per step.
// ---------------------------------------------------------------------------
#define NWG 8
#define HS  (HID / NWG)   // 32 h-indices per WG
#define RPG (4 * HS)      // 128 gate-rows per WG

__global__ void k_lstm(const float* __restrict__ gx, const float* __restrict__ W_hh,
                       float* __restrict__ outputs, unsigned int* __restrict__ sync) {
  extern __shared__ float lds[];
  float* Wl    = lds;                  // [256][RPG] K-major: Wl[k*RPG + r]
  float* hprev = Wl + HID * RPG;       // [2*HID]
  float* gbuf  = hprev + 2 * HID;      // [2][4][HS]
  float* cbuf  = gbuf + 2 * 4 * HS;    // [2][HS]

  const int wg    = blockIdx.x;
  const int tid   = threadIdx.x;       // 0..255
  const int hbase = wg * HS;

  // Load + transpose our W_hh slice into LDS (coalesced on K = tid).
  for (int r = 0; r < RPG; ++r) {
    int gate = r / HS, hh = r % HS;
    int grow = gate * HID + hbase + hh;
    Wl[(size_t)tid * RPG + r] = W_hh[(size_t)grow * HID + tid];
  }
  hprev[tid] = 0.0f;
  hprev[HID + tid] = 0.0f;
  if (tid < 2 * HS) cbuf[tid] = 0.0f;
  __syncthreads();

  const int b    = tid >> 7;        // 0..1
  const int r    = tid & 127;       // gate-row within slice
  const int gate = r / HS;
  const int hh   = r % HS;

  for (int t = 0; t < TT; ++t) {
    // one gate pre-activation per thread: dot(h_prev[b], W_hh[row])
    float acc = gx[(size_t)(b * TT + t) * G4 + gate * HID + hbase + hh];
    const float* hp = hprev + b * HID;
#pragma unroll 4
    for (int k = 0; k < HID; ++k)
      acc = fmaf(hp[k], Wl[(size_t)k * RPG + r], acc);

    float val = (gate == 2) ? tanhf(acc) : (1.0f / (1.0f + expf(-acc)));
    gbuf[(b * 4 + gate) * HS + hh] = val;
    __syncthreads();

    if (tid < 2 * HS) {
      int b2 = tid / HS, h2 = tid % HS;
      float iv = gbuf[(b2 * 4 + 0) * HS + h2];
      float fv = gbuf[(b2 * 4 + 1) * HS + h2];
      float gv = gbuf[(b2 * 4 + 2) * HS + h2];
      float ov = gbuf[(b2 * 4 + 3) * HS + h2];
      float c  = fmaf(fv, cbuf[b2 * HS + h2], iv * gv);
      cbuf[b2 * HS + h2] = c;
      outputs[(size_t)(b2 * TT + t) * HID + hbase + h2] = ov * tanhf(c);
      __threadfence();                 // make h slice visible at device scope
    }
    __syncthreads();

    if (tid == 0) {
      __hip_atomic_fetch_add(sync, 1u, __ATOMIC_RELEASE, __HIP_MEMORY_SCOPE_AGENT);
      unsigned target = (unsigned)NWG * (unsigned)(t + 1);
      while (__hip_atomic_load(sync, __ATOMIC_ACQUIRE, __HIP_MEMORY_SCOPE_AGENT) < target)
        __builtin_amdgcn_s_sleep(1);
    }
    __syncthreads();
    __threadfence();

    // gather the full new hidden state (all WGs' slices) for next step
    hprev[tid]        = outputs[(size_t)(0 * TT + t) * HID + tid];
    hprev[HID + tid]  = outputs[(size_t)(1 * TT + t) * HID + tid];
    __syncthreads();
  }
}

// ---------------------------------------------------------------------------
// Additive attention + causal softmax + context, one wave32 per (b, i) row.
// ---------------------------------------------------------------------------
__global__ void k_attn(const float* __restrict__ aproj, const float* __restrict__ bproj,
                       const float* __restrict__ outputs, const float* __restrict__ V,
                       const float* __restrict__ bV, float* __restrict__ context) {
  __shared__ float srow[8][TT];        // per-wave score row (16 KB)
  int wid = threadIdx.x >> 5, lane = threadIdx.x & 31;
  int m = blockIdx.x * 8 + wid;        // 0..1023
  int b = m / TT, i = m % TT;

  float ai[8], vv[8];
#pragma unroll
  for (int u = 0; u < 8; ++u) {
    ai[u] = aproj[(size_t)m * HID + lane + 32 * u];
    vv[u] = V[lane + 32 * u];
  }
  float bVv = bV[0];

  float mx = -INFINITY;
  for (int j = 0; j <= i; ++j) {
    const float* bj = bproj + (size_t)(b * TT + j) * HID;
    float s = 0.0f;
#pragma unroll
    for (int u = 0; u < 8; ++u)
      s += vv[u] * tanhf(ai[u] + bj[lane + 32 * u]);
    for (int off = 16; off > 0; off >>= 1)
      s += __shfl_xor(s, off, 32);
    s += bVv;                          // all lanes hold full sum now
    if (lane == 0) srow[wid][j] = s;
    mx = fmaxf(mx, s);
  }

  float sum = 0.0f;
  for (int j = lane; j <= i; j += 32) {
    float e = expf(srow[wid][j] - mx);
    srow[wid][j] = e;
    sum += e;
  }
  for (int off = 16; off > 0; off >>= 1)
    sum += __shfl_xor(sum, off, 32);
  float inv = 1.0f / sum;

  float acc[8] = {};
  for (int j = 0; j <= i; ++j) {
    float w = srow[wid][j] * inv;
    const float* oj = outputs + (size_t)(b * TT + j) * HID;
#pragma unroll
    for (int u = 0; u < 8; ++u)
      acc[u] = fmaf(w, oj[lane + 32 * u], acc[u]);
  }
#pragma unroll
  for (int u = 0; u < 8; ++u)
    context[(size_t)m * HID + lane + 32 * u] = acc[u];
}

// ---------------------------------------------------------------------------
// combined = [outputs | context]
// ---------------------------------------------------------------------------
__global__ void k_pack(const float* __restrict__ outputs, const float* __restrict__ context,
                       float* __restrict__ combined) {
  int m = blockIdx.x, h = threadIdx.x;
  combined[(size_t)m * (2 * HID) + h]       = outputs[(size_t)m * HID + h];
  combined[(size_t)m * (2 * HID) + HID + h] = context[(size_t)m * HID + h];
}

__global__ void k_init(unsigned int* sync) { *sync = 0u; }

// ---------------------------------------------------------------------------
extern "C" void kernel_launch(void* const* d_in, const int* in_sizes, int n_in,
                              void* d_out, int out_size, void* d_ws, size_t ws_size,
                              hipStream_t stream) {
  const int*   x    = (const int*)  d_in[0];
  const float* emb  = (const float*)d_in[1];
  const float* W_ih = (const float*)d_in[2];
  const float* W_hh = (const float*)d_in[3];
  const float* b_ih = (const float*)d_in[4];
  const float* b_hh = (const float*)d_in[5];
  const float* W1   = (const float*)d_in[6];
  const float* b1   = (const float*)d_in[7];
  const float* W2   = (const float*)d_in[8];
  const float* b2   = (const float*)d_in[9];
  const float* V    = (const float*)d_in[10];
  const float* bV   = (const float*)d_in[11];
  const float* Wfc  = (const float*)d_in[12];
  const float* bfc  = (const float*)d_in[13];
  float* out = (float*)d_out;

  float* ws = (float*)d_ws;
  float* xe       = ws;                       // 1024*256
  float* gx       = xe + MROWS * EMB;         // 1024*1024
  float* outputs  = gx + MROWS * G4;          // 1024*256
  float* aproj    = outputs + MROWS * HID;    // 1024*256
  float* bproj    = aproj + MROWS * HID;      // 1024*256
  float* context  = bproj + MROWS * HID;      // 1024*256
  float* combined = context + MROWS * HID;    // 1024*512
  unsigned int* sync = (unsigned int*)(combined + MROWS * 2 * HID);

  k_init<<<1, 1, 0, stream>>>(sync);
  k_embed<<<MROWS, EMB, 0, stream>>>(x, emb, xe);

  // gx = xe @ W_ih^T + b_ih + b_hh   (M=1024, N=1024, K=256)
  {
    int tiles = (MROWS / (16 * MBLK)) * (G4 / (16 * NBLK));   // 16*32 = 512
    k_gemm_f32_wmma<<<tiles / 8, 256, 0, stream>>>(xe, W_ih, b_ih, b_hh, gx,
                                                   MROWS, G4, EMB);
  }

  // sequential LSTM, 8 cooperating workgroups, W_hh resident in LDS
  {
    size_t ldsz = (size_t)(HID * RPG + 2 * HID + 2 * 4 * HS + 2 * HS) * sizeof(float);
    k_lstm<<<NWG, 256, ldsz, stream>>>(gx, W_hh, outputs, sync);
  }

  // a = outputs @ W1^T + b1 ; b = outputs @ W2^T + b2  (M=1024, N=256, K=256)
  {
    int tiles = (MROWS / (16 * MBLK)) * (HID / (16 * NBLK));  // 16*8 = 128
    k_gemm_f32_wmma<<<tiles / 8, 256, 0, stream>>>(outputs, W1, b1, nullptr, aproj,
                                                   MROWS, HID, HID);
    k_gemm_f32_wmma<<<tiles / 8, 256, 0, stream>>>(outputs, W2, b2, nullptr, bproj,
                                                   MROWS, HID, HID);
  }

  k_attn<<<MROWS / 8, 256, 0, stream>>>(aproj, bproj, outputs, V, bV, context);
  k_pack<<<MROWS, HID, 0, stream>>>(outputs, context, combined);

  // logits = combined @ Wfc^T + bfc  (M=1024, N=32000, K=512)
  {
    int tiles = (MROWS / (16 * MBLK)) * (VOCAB / (16 * NBLK));  // 16*1000 = 16000
    k_gemm_f32_wmma<<<tiles / 8, 256, 0, stream>>>(combined, Wfc, bfc, nullptr, out,
                                                   MROWS, VOCAB, 2 * HID);
  }
}